// DiscourseGNN_46316927320307
// MI455X (gfx1250) — compile-verified
//
#include <hip/hip_runtime.h>
#include <hip/hip_bf16.h>
#include <math.h>

// ---------------------------------------------------------------------------
// DiscourseGNN on MI455X (gfx1250): fused GAT with sparse-softmax reduction.
// GEMMs run on V_WMMA_F32_16X16X4_F32 (fp32 matches the fp32 reference).
// ---------------------------------------------------------------------------

#define NN   8192
#define DIN  512
#define HH   256
#define NL   2
#define NE   262144
#define LN_EPS 1e-5f

typedef float v2f __attribute__((ext_vector_type(2)));
typedef float v8f __attribute__((ext_vector_type(8)));

__device__ __forceinline__ v8f wmma_f32_k4(v2f a, v2f b, v8f c) {
    // D = A(16x4) x B(4x16) + C(16x16), fp32, wave32
    return __builtin_amdgcn_wmma_f32_16x16x4_f32(
        /*neg_a=*/false, a, /*neg_b=*/false, b,
        /*c_mod=*/(short)0, c, /*reuse_a=*/false, /*reuse_b=*/false);
}

// Order-preserving float<->uint mapping so atomicMax(uint) == float max.
__device__ __forceinline__ unsigned f2ord(float f) {
    unsigned u = __float_as_uint(f);
    return (u & 0x80000000u) ? ~u : (u | 0x80000000u);
}
__device__ __forceinline__ float ord2f(unsigned k) {
    unsigned u = (k & 0x80000000u) ? (k & 0x7fffffffu) : ~k;
    return __uint_as_float(u);
}

// ---------------------------------------------------------------------------
// GEMM: C[M,N] = A[M,K] @ B[N,K]^T (+ optional bias[N]); row-major fp32.
// One wave computes a 32x32 tile: 4 accumulators, 4 wmma per K-step of 4.
// ---------------------------------------------------------------------------
__global__ void gemm_wmma_f32(const float* __restrict__ A,
                              const float* __restrict__ B,
                              const float* __restrict__ bias,
                              float* __restrict__ C,
                              int M, int N, int K)
{
    const int gid   = blockIdx.x * blockDim.x + threadIdx.x;
    const int wid   = gid >> 5;
    const int lane  = threadIdx.x & 31;
    const int tilesN = N >> 5;
    const int tm = (wid / tilesN) << 5;
    const int tn = (wid % tilesN) << 5;
    if (tm >= M) return;

    const int r  = lane & 15;          // row/col within 16-slice
    const int kb = (lane >> 4) << 1;   // K sub-offset: 0 (lanes 0-15) / 2 (lanes 16-31)

    const float* Ap0 = A + (size_t)(tm + r)      * K + kb;
    const float* Ap1 = A + (size_t)(tm + 16 + r) * K + kb;
    const float* Bp0 = B + (size_t)(tn + r)      * K + kb;
    const float* Bp1 = B + (size_t)(tn + 16 + r) * K + kb;

    v8f acc00 = {}, acc01 = {}, acc10 = {}, acc11 = {};

    #pragma unroll 2
    for (int k = 0; k < K; k += 4) {
        v2f a0 = *(const v2f*)(Ap0 + k);
        v2f a1 = *(const v2f*)(Ap1 + k);
        v2f b0 = *(const v2f*)(Bp0 + k);
        v2f b1 = *(const v2f*)(Bp1 + k);
        if (k + 32 < K) {   // gfx1250: global_prefetch_b8
            __builtin_prefetch(Ap0 + k + 32, 0, 3);
            __builtin_prefetch(Bp0 + k + 32, 0, 3);
        }
        acc00 = wmma_f32_k4(a0, b0, acc00);
        acc01 = wmma_f32_k4(a0, b1, acc01);
        acc10 = wmma_f32_k4(a1, b0, acc10);
        acc11 = wmma_f32_k4(a1, b1, acc11);
    }

    // C/D layout: VGPR v -> row v (lanes 0-15) / row 8+v (lanes 16-31), col = lane&15
    const int hi = (lane >> 4) << 3;
    const int c0 = tn + r, c1 = tn + 16 + r;
    const float bv0 = bias ? bias[c0] : 0.f;
    const float bv1 = bias ? bias[c1] : 0.f;
    #pragma unroll
    for (int v = 0; v < 8; ++v) {
        const int r0 = tm + hi + v;
        const int r1 = tm + 16 + hi + v;
        C[(size_t)r0 * N + c0] = acc00[v] + bv0;
        C[(size_t)r0 * N + c1] = acc01[v] + bv1;
        C[(size_t)r1 * N + c0] = acc10[v] + bv0;
        C[(size_t)r1 * N + c1] = acc11[v] + bv1;
    }
}

// ---------------------------------------------------------------------------
// Per-node attention scores: s_src[i] = hw[i]·a[:H], s_tgt[i] = hw[i]·a[H:]
// One wave (32 lanes) per node.
// ---------------------------------------------------------------------------
__global__ void node_scores(const float* __restrict__ hw,
                            const float* __restrict__ a_lo,
                            const float* __restrict__ a_hi,
                            float* __restrict__ s_src,
                            float* __restrict__ s_tgt)
{
    const int gid  = blockIdx.x * blockDim.x + threadIdx.x;
    const int node = gid >> 5;
    const int lane = threadIdx.x & 31;
    if (node >= NN) return;
    const float* row = hw + (size_t)node * HH;
    float sa = 0.f, st = 0.f;
    #pragma unroll
    for (int c = lane; c < HH; c += 32) {
        const float v = row[c];
        sa += v * a_lo[c];
        st += v * a_hi[c];
    }
    #pragma unroll
    for (int off = 16; off > 0; off >>= 1) {
        sa += __shfl_xor(sa, off, 32);
        st += __shfl_xor(st, off, 32);
    }
    if (lane == 0) { s_src[node] = sa; s_tgt[node] = st; }
}

// Column sums S[c] = sum_i hw[i,c]  (one block covers 64 rows, one atomic/col)
#define CS_ROWS 64
__global__ void col_sum(const float* __restrict__ hw, float* __restrict__ S)
{
    const int col = threadIdx.x;
    const int r0  = blockIdx.x * CS_ROWS;
    float acc = 0.f;
    for (int r = r0; r < r0 + CS_ROWS; ++r)
        acc += hw[(size_t)r * HH + col];
    atomicAdd(&S[col], acc);
}

// Zero / re-init per-layer accumulators (graph-replay safe: runs every call).
__global__ void init_layer(float* __restrict__ acc, float* __restrict__ Zacc,
                           unsigned* __restrict__ mkey, float* __restrict__ S)
{
    const int tid    = blockIdx.x * blockDim.x + threadIdx.x;
    const int stride = gridDim.x * blockDim.x;
    for (int j = tid; j < NN * HH; j += stride) acc[j] = 0.f;
    for (int j = tid; j < NN; j += stride) { Zacc[j] = 0.f; mkey[j] = f2ord(0.f); }
    for (int j = tid; j < HH; j += stride) S[j] = 0.f;
}

// Pass 1: edge logit + LeakyReLU; track per-src row max (vs implicit zeros).
__global__ void edge_logits(const int* __restrict__ ei,
                            const float* __restrict__ s_src,
                            const float* __restrict__ s_tgt,
                            float* __restrict__ ebuf,
                            unsigned* __restrict__ mkey)
{
    const int e = blockIdx.x * blockDim.x + threadIdx.x;
    if (e >= NE) return;
    const int src = ei[e];
    const int tgt = ei[NE + e];
    float v = s_src[src] + s_tgt[tgt];
    v = (v > 0.f) ? v : 0.2f * v;
    ebuf[e] = v;
    atomicMax(&mkey[src], f2ord(v));
}

// Pass 2: coefficient c_e = exp(e - m_src) - exp(-m_src); accumulate Z.
__global__ void edge_coefs(const int* __restrict__ ei,
                           const unsigned* __restrict__ mkey,
                           float* __restrict__ ebuf,
                           float* __restrict__ Zacc)
{
    const int e = blockIdx.x * blockDim.x + threadIdx.x;
    if (e >= NE) return;
    const int src = ei[e];
    const float m = ord2f(mkey[src]);
    const float c = expf(ebuf[e] - m) - expf(-m);
    ebuf[e] = c;
    atomicAdd(&Zacc[src], c);
}

// Pass 3: scatter numerator acc[src,:] += c_e * hw[tgt,:]  (one wave / edge)
__global__ void edge_scatter(const int* __restrict__ ei,
                             const float* __restrict__ ebuf,
                             const float* __restrict__ hw,
                             float* __restrict__ acc)
{
    const int gid  = blockIdx.x * blockDim.x + threadIdx.x;
    const int e    = gid >> 5;
    const int lane = threadIdx.x & 31;
    if (e >= NE) return;
    const int src = ei[e];
    const int tgt = ei[NE + e];
    const float c = ebuf[e];
    const float* hrow = hw + (size_t)tgt * HH;
    float* arow = acc + (size_t)src * HH;
    #pragma unroll
    for (int col = lane; col < HH; col += 32)
        atomicAdd(&arow[col], c * hrow[col]);
}

// Pass 4: h = elu(layernorm(h + h_new))   (one 256-thread block per node)
__global__ void finalize_node(const unsigned* __restrict__ mkey,
                              const float* __restrict__ Zacc,
                              const float* __restrict__ acc,
                              const float* __restrict__ S,
                              const float* __restrict__ g,
                              const float* __restrict__ b,
                              float* __restrict__ h)
{
    const int i = blockIdx.x;
    const int t = threadIdx.x;
    const float m  = ord2f(mkey[i]);
    const float em = expf(-m);
    const float Z  = Zacc[i] + (float)NN * em;
    const float hn = (acc[(size_t)i * HH + t] + em * S[t]) / Z;
    const float x  = h[(size_t)i * HH + t] + hn;

    __shared__ float red[HH];
    red[t] = x;
    __syncthreads();
    #pragma unroll
    for (int s = HH / 2; s > 0; s >>= 1) {
        if (t < s) red[t] += red[t + s];
        __syncthreads();
    }
    const float mu = red[0] * (1.0f / HH);
    __syncthreads();
    const float d = x - mu;
    red[t] = d * d;
    __syncthreads();
    #pragma unroll
    for (int s = HH / 2; s > 0; s >>= 1) {
        if (t < s) red[t] += red[t + s];
        __syncthreads();
    }
    const float var = red[0] * (1.0f / HH);
    const float y = d * rsqrtf(var + LN_EPS) * g[t] + b[t];
    h[(size_t)i * HH + t] = (y > 0.f) ? y : expm1f(y);
}

__global__ void zero_gemb(float* __restrict__ gemb)
{
    gemb[threadIdx.x] = 0.f;
}

// gate = sigmoid(h·W_pool + b); out_h = h; graph_emb += gate*h  (wave / node)
__global__ void pool_out(const float* __restrict__ h,
                         const float* __restrict__ Wp,
                         const float* __restrict__ bp,
                         float* __restrict__ out_h,
                         float* __restrict__ gemb)
{
    const int gid  = blockIdx.x * blockDim.x + threadIdx.x;
    const int node = gid >> 5;
    const int lane = threadIdx.x & 31;
    if (node >= NN) return;
    const float* hrow = h + (size_t)node * HH;
    float dot = 0.f;
    #pragma unroll
    for (int c = lane; c < HH; c += 32) dot += hrow[c] * Wp[c];
    #pragma unroll
    for (int off = 16; off > 0; off >>= 1) dot += __shfl_xor(dot, off, 32);
    const float gate = 1.0f / (1.0f + expf(-(dot + bp[0])));
    #pragma unroll
    for (int c = lane; c < HH; c += 32) {
        const float v = hrow[c];
        out_h[(size_t)node * HH + c] = v;
        atomicAdd(&gemb[c], gate * v);
    }
}

// ---------------------------------------------------------------------------
extern "C" void kernel_launch(void* const* d_in, const int* in_sizes, int n_in,
                              void* d_out, int out_size, void* d_ws, size_t ws_size,
                              hipStream_t stream)
{
    (void)in_sizes; (void)n_in; (void)out_size; (void)ws_size;

    const float* X    = (const float*)d_in[0];
    const int*   EI   = (const int*)  d_in[1];
    const float* Win  = (const float*)d_in[2];
    const float* bin  = (const float*)d_in[3];
    const float* Wg   = (const float*)d_in[4];   // (L,H,H)
    const float* ag   = (const float*)d_in[5];   // (L,2H)
    const float* lng  = (const float*)d_in[6];
    const float* lnb  = (const float*)d_in[7];
    const float* Wp   = (const float*)d_in[8];
    const float* bp   = (const float*)d_in[9];
    float* out = (float*)d_out;                  // [N*H] h  +  [H] graph_emb

    // Workspace layout (fp32)
    char* ws = (char*)d_ws;
    const size_t NH = (size_t)NN * HH;
    float*    h     = (float*)ws;                         ws += NH * 4;
    float*    hw    = (float*)ws;                         ws += NH * 4;
    float*    acc   = (float*)ws;                         ws += NH * 4;
    float*    ebuf  = (float*)ws;                         ws += (size_t)NE * 4;
    float*    ssrc  = (float*)ws;                         ws += (size_t)NN * 4;
    float*    stgt  = (float*)ws;                         ws += (size_t)NN * 4;
    float*    Zacc  = (float*)ws;                         ws += (size_t)NN * 4;
    float*    S     = (float*)ws;                         ws += (size_t)HH * 4;
    unsigned* mkey  = (unsigned*)ws;                      ws += (size_t)NN * 4;

    const int TB = 256;

    // h = X @ W_in^T + b_in   (M=8192, N=256, K=512)
    {
        const int waves  = (NN / 32) * (HH / 32);
        const int blocks = waves * 32 / TB;
        gemm_wmma_f32<<<blocks, TB, 0, stream>>>(X, Win, bin, h, NN, HH, DIN);
    }

    for (int l = 0; l < NL; ++l) {
        const float* Wgl = Wg + (size_t)l * HH * HH;
        const float* alo = ag + (size_t)l * 2 * HH;
        const float* ahi = alo + HH;

        // hw = h @ W_gat[l]^T   (M=8192, N=256, K=256)
        {
            const int waves  = (NN / 32) * (HH / 32);
            const int blocks = waves * 32 / TB;
            gemm_wmma_f32<<<blocks, TB, 0, stream>>>(h, Wgl, nullptr, hw, NN, HH, HH);
        }

        init_layer<<<2048, TB, 0, stream>>>(acc, Zacc, mkey, S);
        node_scores<<<NN * 32 / TB, TB, 0, stream>>>(hw, alo, ahi, ssrc, stgt);
        col_sum<<<NN / CS_ROWS, HH, 0, stream>>>(hw, S);
        edge_logits<<<NE / TB, TB, 0, stream>>>(EI, ssrc, stgt, ebuf, mkey);
        edge_coefs<<<NE / TB, TB, 0, stream>>>(EI, mkey, ebuf, Zacc);
        edge_scatter<<<(size_t)NE * 32 / TB, TB, 0, stream>>>(EI, ebuf, hw, acc);
        finalize_node<<<NN, HH, 0, stream>>>(mkey, Zacc, acc, S,
                                             lng + (size_t)l * HH,
                                             lnb + (size_t)l * HH, h);
    }

    zero_gemb<<<1, HH, 0, stream>>>(out + NH);
    pool_out<<<NN * 32 / TB, TB, 0, stream>>>(h, Wp, bp, out, out + NH);
}